// HeteroMLPPredictor_10720238371561
// MI455X (gfx1250) — compile-verified
//
#include <hip/hip_runtime.h>
#include <hip/hip_bf16.h>

typedef __attribute__((ext_vector_type(2))) float v2f;
typedef __attribute__((ext_vector_type(8))) float v8f;

// ---------------------------------------------------------------------------
// Kernel 1: per-node projection P = h @ Weff^T using V_WMMA_F32_16X16X4_F32.
//   One wave (32 lanes) per 16-node tile. D(16x16) += A(16x4) x B(4x16),
//   32 K-steps covering K=128.
//   Weff columns: n=0..2 -> W[n][0:128] (src half), n=3..5 -> W[n-3][128:256]
//   (dst half). Columns 6..15 load dummy (in-bounds) data from W; since
//   B column n only contributes to D column n, and D columns 6..15 are never
//   stored, no masking is needed -> WMMA feeds straight from the loads.
//   P layout (stride 8 floats per node): [0..2]=src proj, [4..6]=dst proj.
// ---------------------------------------------------------------------------
__global__ __launch_bounds__(32) void gnn_proj_wmma(
    const float* __restrict__ h,      // (nodes, 128)
    const float* __restrict__ W,      // (3, 256)
    float* __restrict__ P)            // (nodes, 8) scratch
{
    const int tile = blockIdx.x;        // tile of 16 nodes
    const int lane = threadIdx.x;       // 0..31
    const int rc   = lane & 15;         // A-row / B-col index
    const int koff = (lane >> 4) * 2;   // K sub-offset: 0 or 2

    // A source: row (tile*16 + rc) of h, starting at K sub-offset.
    const float* hrow = h + (size_t)(tile * 16 + rc) * 128 + koff;

    // B source: effective-weight row for column rc.
    // rc>=6: harmless in-bounds dummy reads (their D columns are discarded).
    const float* wrow = W + koff;                            // dummy (rc >= 6)
    if (rc < 3)      wrow = W + rc * 256 + koff;             // src half
    else if (rc < 6) wrow = W + (rc - 3) * 256 + 128 + koff; // dst half

    v8f acc = {};
#pragma unroll
    for (int kk = 0; kk < 32; ++kk) {
        v2f a = *(const v2f*)(hrow + kk * 4);
        v2f b = *(const v2f*)(wrow + kk * 4);
        // 8 args: (neg_a, A, neg_b, B, c_mod, C, reuse_a, reuse_b)
        acc = __builtin_amdgcn_wmma_f32_16x16x4_f32(
            false, a, false, b, (short)0, acc, false, false);
    }

    // D layout: VGPR r, lanes 0-15: M=r,N=lane; lanes 16-31: M=r+8,N=lane-16.
    if (rc < 6) {
        const int colOff = (rc < 3) ? rc : (rc + 1);        // 0,1,2, 4,5,6
        const int mbase  = tile * 16 + ((lane >> 4) * 8);
#pragma unroll
        for (int r = 0; r < 8; ++r) {
            P[(size_t)(mbase + r) * 8 + colOff] = acc[r];
        }
    }
}

// ---------------------------------------------------------------------------
// Kernel 2: per-edge gather-add. out[e,c] = P[src[e]][c] + P[dst[e]][4+c] + b[c]
// Two aligned b128 loads from L2-resident P, three coalesced b32 stores.
// ---------------------------------------------------------------------------
__global__ __launch_bounds__(256) void gnn_edge_gather(
    const int* __restrict__ src,
    const int* __restrict__ dst,
    const float* __restrict__ P,
    const float* __restrict__ b,
    float* __restrict__ out,
    int n_edges)
{
    const int e = blockIdx.x * blockDim.x + threadIdx.x;
    if (e >= n_edges) return;

    const int s = src[e];
    const int d = dst[e];

    const float4 ps = *(const float4*)(P + (size_t)s * 8);      // src proj
    const float4 pd = *(const float4*)(P + (size_t)d * 8 + 4);  // dst proj

    const float b0 = b[0], b1 = b[1], b2 = b[2];   // uniform -> scalar loads

    float* o = out + (size_t)e * 3;
    o[0] = ps.x + pd.x + b0;
    o[1] = ps.y + pd.y + b1;
    o[2] = ps.z + pd.z + b2;
}

// ---------------------------------------------------------------------------
// Inputs (setup_inputs order): 0:h (nodes*128 f32), 1:src (E int), 2:dst (E int),
//                              3:W (3*256 f32), 4:b (3 f32)
// Output: (E, 3) f32. Workspace: P = nodes*8 f32 (320 KB for 10000 nodes).
// ---------------------------------------------------------------------------
extern "C" void kernel_launch(void* const* d_in, const int* in_sizes, int n_in,
                              void* d_out, int out_size, void* d_ws, size_t ws_size,
                              hipStream_t stream) {
    const float* h   = (const float*)d_in[0];
    const int*   src = (const int*)d_in[1];
    const int*   dst = (const int*)d_in[2];
    const float* W   = (const float*)d_in[3];
    const float* b   = (const float*)d_in[4];
    float* out = (float*)d_out;

    const int n_nodes = in_sizes[0] / 128;   // 10000
    const int n_edges = in_sizes[1];         // 500000

    float* P = (float*)d_ws;                 // n_nodes * 8 floats

    const int n_tiles = (n_nodes + 15) / 16; // 625 (exact)
    gnn_proj_wmma<<<n_tiles, 32, 0, stream>>>(h, W, P);

    const int threads = 256;
    const int blocks  = (n_edges + threads - 1) / threads;
    gnn_edge_gather<<<blocks, threads, 0, stream>>>(src, dst, P, b, out, n_edges);
}